// CRF_16913581211906
// MI455X (gfx1250) — compile-verified
//
#include <hip/hip_runtime.h>
#include <cstdint>
#include <cstddef>

typedef float v2f __attribute__((ext_vector_type(2)));
typedef float v8f __attribute__((ext_vector_type(8)));

#define S_LEN 256
#define NLAB  17
#define LTOT  19   // 17 labels + start(17) + end(18)

// lane n <-> lane n^16 full exchange (group-of-32 swizzle, xor_mask=0x10, and=0x1f)
__device__ __forceinline__ float swz16(float x) {
  return __int_as_float(__builtin_amdgcn_ds_swizzle(__float_as_int(x), 0x401F));
}

// ---------------------------------------------------------------------------
// Gold (numerator) score: unary + transition terms along the labeled path.
// One block per sequence, lanes parallel over time, LDS tree reduce.
// ---------------------------------------------------------------------------
__global__ __launch_bounds__(256) void crf_gold_kernel(
    const float* __restrict__ logits, const int* __restrict__ labels,
    const int* __restrict__ lens, const float* __restrict__ T,
    float* __restrict__ gold) {
  const int b = blockIdx.x;
  const int t = threadIdx.x;
  __shared__ float red[256];

  const int len = lens[b];
  const int* lab = labels + (size_t)b * S_LEN;

  float v = 0.0f;
  if (t < len) {
    const int lt = lab[t];
    v = logits[((size_t)b * S_LEN + t) * NLAB + lt];       // unary
    const int prev = (t == 0) ? (LTOT - 2) : lab[t - 1];   // start at t==0
    v += T[lt * LTOT + prev];                              // T[to, from]
    if (t == len - 1) v += T[(LTOT - 1) * LTOT + lt];      // -> end
  }
  red[t] = v;
  __syncthreads();
#pragma unroll
  for (int off = 128; off > 0; off >>= 1) {
    if (t < off) red[t] += red[t + off];
    __syncthreads();
  }
  if (t == 0) gold[b] = red[0];
}

// ---------------------------------------------------------------------------
// Partition function via scaled linear-space forward algorithm on WMMA.
// One wave per block; each wave owns TWO independent 16-column batch tiles
// for ILP inside the serial time recursion.
//   state_t(19 x 16) = ( exp(T)(19x19) @ state_{t-1} ) .* exp(logit_t)
// Rows 0..15 via V_WMMA_F32_16X16X4_F32 (5 K-chunks); row 16 via a per-lane
// VALU dot product (rows 17/18 are identically zero after step 0 because
// their emissions are exp(-1e4)=0). Per-column rescale keeps f32 in range.
// ---------------------------------------------------------------------------
__global__ __launch_bounds__(32) void crf_forward_kernel(
    const float* __restrict__ logits, const int* __restrict__ lens,
    const float* __restrict__ T, const float* __restrict__ gold,
    float* __restrict__ out) {
  const int lane = threadIdx.x;      // 0..31
  const int half = lane >> 4;        // 0: lanes 0-15, 1: lanes 16-31
  const int col  = lane & 15;

  // --- exp(T) rows 0..15 in WMMA-A layout (K padded 19 -> 20), and
  //     per-lane row-16 weights + end-row weights for the epilogue ----------
  v2f   MA0[5];
  float W16[5][2];   // exp(T[16, j]) for this lane's j's
  float WEND[5][2];  // exp(T[18, j]) for this lane's j's
#pragma unroll
  for (int kc = 0; kc < 5; ++kc) {
    float a0[2];
#pragma unroll
    for (int r = 0; r < 2; ++r) {
      const int k = kc * 4 + r + 2 * half;   // "from" index j
      a0[r]       = (k < LTOT) ? __expf(T[col * LTOT + k]) : 0.0f;
      W16[kc][r]  = (k < LTOT) ? __expf(T[16 * LTOT + k]) : 0.0f;
      WEND[kc][r] = (k < LTOT) ? __expf(T[(LTOT - 1) * LTOT + k]) : 0.0f;
    }
    MA0[kc] = {a0[0], a0[1]};
  }

  // --- per-tile state (B-layout), log-scale accumulators --------------------
  int   bidx[2], lenv[2];
  const float* lbase[2];
  float Bs[2][5][2];
  float acc[2];
#pragma unroll
  for (int u = 0; u < 2; ++u) {
    bidx[u]  = blockIdx.x * 32 + u * 16 + col;
    lenv[u]  = lens[bidx[u]];
    lbase[u] = logits + (size_t)bidx[u] * S_LEN * NLAB;
    acc[u]   = 0.0f;
#pragma unroll
    for (int kc = 0; kc < 5; ++kc) { Bs[u][kc][0] = 0.0f; Bs[u][kc][1] = 0.0f; }
    if (half == 0) Bs[u][4][1] = 1.0f;   // alpha0[start=17] = 1 (j=17: kc=4,r=1,half=0)
  }

#pragma unroll 2
  for (int t = 0; t < S_LEN; ++t) {
#pragma unroll
    for (int u = 0; u < 2; ++u) {
      const float* lp = lbase[u] + t * NLAB;
      __builtin_prefetch(lp + 2 * NLAB, 0, 0);   // global_prefetch_b8, 2 steps ahead

      // emission factors: low lanes rows 0..7, high lanes rows 8..15, plus row 16
      float e[8];
#pragma unroll
      for (int r = 0; r < 8; ++r) e[r] = __expf(lp[half * 8 + r]);
      const float e16 = __expf(lp[16]);

      // rows 0..15:  nxt = exp(T)[0:16,:] @ state   (five K-chunks)
      v8f c0 = {};
#pragma unroll
      for (int kc = 0; kc < 5; ++kc) {
        v2f bv = {Bs[u][kc][0], Bs[u][kc][1]};
        c0 = __builtin_amdgcn_wmma_f32_16x16x4_f32(false, MA0[kc], false, bv,
                                                   (short)0, c0, false, false);
      }

      // row 16: per-lane partial dot over this lane's 10 j's, combine halves
      float p16 = 0.0f;
#pragma unroll
      for (int kc = 0; kc < 5; ++kc) {
        p16 = fmaf(Bs[u][kc][0], W16[kc][0], p16);
        p16 = fmaf(Bs[u][kc][1], W16[kc][1], p16);
      }
      const float row16 = p16 + swz16(p16);

      // multiply by exp(logits)
      float d[8];
#pragma unroll
      for (int r = 0; r < 8; ++r) d[r] = c0[r] * e[r];
      const float r16 = (half == 0) ? (row16 * e16) : 0.0f;

      // C-layout -> B-layout: lane<->lane^16 exchanges
      const float s0 = swz16(d[0]), s1 = swz16(d[1]), s2 = swz16(d[2]), s3 = swz16(d[3]);
      const float s4 = swz16(d[4]), s5 = swz16(d[5]), s6 = swz16(d[6]), s7 = swz16(d[7]);
      float nb[5][2];
      nb[0][0] = half ? s2   : d[0];  nb[0][1] = half ? s3   : d[1];   // j=0,1 | 2,3
      nb[1][0] = half ? s6   : d[4];  nb[1][1] = half ? s7   : d[5];   // j=4,5 | 6,7
      nb[2][0] = half ? d[2] : s0;    nb[2][1] = half ? d[3] : s1;     // j=8,9 |10,11
      nb[3][0] = half ? d[6] : s4;    nb[3][1] = half ? d[7] : s5;     // j=12,13|14,15
      nb[4][0] = r16;                 nb[4][1] = 0.0f;                 // j=16,17|18,19

      // sequence mask: keep previous alpha for t >= len
      const bool act = (t < lenv[u]);
#pragma unroll
      for (int kc = 0; kc < 5; ++kc) {
        Bs[u][kc][0] = act ? nb[kc][0] : Bs[u][kc][0];
        Bs[u][kc][1] = act ? nb[kc][1] : Bs[u][kc][1];
      }

      // per-column rescale (invariant-preserving for masked columns too)
      float m = Bs[u][0][0];
#pragma unroll
      for (int kc = 0; kc < 5; ++kc) {
        m = fmaxf(m, Bs[u][kc][0]);
        m = fmaxf(m, Bs[u][kc][1]);
      }
      m = fmaxf(m, swz16(m));        // combine both lane halves of the column
      m = fmaxf(m, 1e-30f);
      const float inv = 1.0f / m;
      acc[u] += __logf(m);
#pragma unroll
      for (int kc = 0; kc < 5; ++kc) {
        Bs[u][kc][0] *= inv;
        Bs[u][kc][1] *= inv;
      }
    }
  }

  // norm = acc + log( sum_j state_j * exp(T[end, j]) );  out = gold - norm
#pragma unroll
  for (int u = 0; u < 2; ++u) {
    float part = 0.0f;
#pragma unroll
    for (int kc = 0; kc < 5; ++kc) {
      part = fmaf(Bs[u][kc][0], WEND[kc][0], part);
      part = fmaf(Bs[u][kc][1], WEND[kc][1], part);
    }
    const float tot  = part + swz16(part);
    const float norm = acc[u] + __logf(tot);
    if (half == 0) out[bidx[u]] = gold[bidx[u]] - norm;
  }
}

// ---------------------------------------------------------------------------
extern "C" void kernel_launch(void* const* d_in, const int* in_sizes, int n_in,
                              void* d_out, int out_size, void* d_ws, size_t ws_size,
                              hipStream_t stream) {
  const float* logits = (const float*)d_in[0];
  const int*   labels = (const int*)d_in[1];
  const int*   lens   = (const int*)d_in[2];
  const float* T      = (const float*)d_in[3];
  float* out  = (float*)d_out;
  float* gold = (float*)d_ws;          // B floats of scratch

  const int B = in_sizes[2];           // 4096

  crf_gold_kernel<<<B, 256, 0, stream>>>(logits, labels, lens, T, gold);
  crf_forward_kernel<<<B / 32, 32, 0, stream>>>(logits, lens, T, gold, out);
}